// BiasedLoss_15393162789369
// MI455X (gfx1250) — compile-verified
//
#include <hip/hip_runtime.h>

typedef __attribute__((ext_vector_type(4))) float v4f;
typedef __attribute__((ext_vector_type(2))) float v2f;
typedef __attribute__((ext_vector_type(8))) float v8f;

constexpr int kN      = 262144;
constexpr int kC      = 128;
constexpr int kTPB    = 256;                   // 8 wave32 per block
constexpr int kBlocks = 2048;                  // 16384 waves total -> 16 rows/wave
constexpr int kWaves  = kBlocks * (kTPB / 32);

// Pass 1: one wave per row. Coalesced b128 NT loads, wave argmax (first-max
// tie-break like jnp.argmax), conditional mask, per-lane |x*t| accumulation
// across rows, then wave+block reduction to one partial per block.
__global__ __launch_bounds__(kTPB) void biased_loss_partial(
    const float* __restrict__ x, const float* __restrict__ tgt,
    float* __restrict__ partial) {
  const int lane        = threadIdx.x & 31;
  const int waveInBlock = threadIdx.x >> 5;
  const int gwave       = blockIdx.x * (kTPB >> 5) + waveInBlock;
  const int base        = lane << 2;           // first column this lane owns

  float acc = 0.0f;
  for (int row = gwave; row < kN; row += kWaves) {
    const size_t rb = (size_t)row * kC + (size_t)base;
    v4f xv = __builtin_nontemporal_load((const v4f*)(x + rb));
    v4f tv = __builtin_nontemporal_load((const v4f*)(tgt + rb));

    // Lane-local argmax over 4 elements; strict '>' keeps the earliest index.
    float m = xv.x; int mi = base;
    if (xv.y > m) { m = xv.y; mi = base + 1; }
    if (xv.z > m) { m = xv.z; mi = base + 2; }
    if (xv.w > m) { m = xv.w; mi = base + 3; }

    // Wave32 butterfly argmax: (max value, lowest index among maxima).
#pragma unroll
    for (int off = 16; off > 0; off >>= 1) {
      float om = __shfl_xor(m, off, 32);
      int   oi = __shfl_xor(mi, off, 32);
      if (om > m || (om == m && oi < mi)) { m = om; mi = oi; }
    }

    // cond = (argmax > 0) && (target[row][0] == 0)
    const float t0   = __shfl(tv.x, 0, 32);
    const bool  cond = (mi > 0) && (t0 == 0.0f);
    if (cond) {
      tv.x = (base + 0 == mi) ? tv.x : 0.0f;
      tv.y = (base + 1 == mi) ? tv.y : 0.0f;
      tv.z = (base + 2 == mi) ? tv.z : 0.0f;
      tv.w = (base + 3 == mi) ? tv.w : 0.0f;
    }
    acc += __builtin_fabsf(xv.x * tv.x) + __builtin_fabsf(xv.y * tv.y) +
           __builtin_fabsf(xv.z * tv.z) + __builtin_fabsf(xv.w * tv.w);
  }

  // Wave sum, then deterministic serial block sum (8 values).
#pragma unroll
  for (int off = 16; off > 0; off >>= 1) acc += __shfl_xor(acc, off, 32);

  __shared__ float smem[kTPB / 32];
  if (lane == 0) smem[waveInBlock] = acc;
  __syncthreads();
  if (threadIdx.x == 0) {
    float s = 0.0f;
#pragma unroll
    for (int i = 0; i < kTPB / 32; ++i) s += smem[i];
    partial[blockIdx.x] = s;
  }
}

// Pass 2: single wave. Reduce the 2048 block partials via
// V_WMMA_F32_16X16X4_F32 with an all-ones B matrix:
//   D[m][n] += sum_k A[m][k]  (every column of C accumulates the row sums)
// Each WMMA consumes 64 partials (2 per lane in the 16x4 A tile). After the
// loop, column 0's 16 entries live in c[0..7] of lanes 0 and 16; summing the
// 8 regs per lane and adding across the half-waves yields the grand total.
// Deterministic (fixed association), full-EXEC (blockDim == 32).
__global__ __launch_bounds__(32) void biased_loss_final(
    const float* __restrict__ partial, float* __restrict__ out) {
  const int lane = threadIdx.x;

  v8f c = {};
  v2f ones; ones.x = 1.0f; ones.y = 1.0f;

  for (int b = 0; b < kBlocks; b += 64) {
    v2f a;
    a.x = partial[b + 2 * lane];
    a.y = partial[b + 2 * lane + 1];
    // 8 args: (neg_a, A, neg_b, B, c_mod, C, reuse_a, reuse_b)
    c = __builtin_amdgcn_wmma_f32_16x16x4_f32(
        false, a, false, ones, (short)0, c, false, false);
  }

  float s = c[0] + c[1] + c[2] + c[3] + c[4] + c[5] + c[6] + c[7];
  s += __shfl_xor(s, 16, 32);
  if (lane == 0) out[0] = s * (1.0f / 33554432.0f);  // / (N*C) = 2^-25, exact
}

extern "C" void kernel_launch(void* const* d_in, const int* in_sizes, int n_in,
                              void* d_out, int out_size, void* d_ws, size_t ws_size,
                              hipStream_t stream) {
  const float* x   = (const float*)d_in[0];
  const float* tgt = (const float*)d_in[1];
  float* out       = (float*)d_out;
  float* partials  = (float*)d_ws;  // kBlocks floats = 8 KiB of scratch

  biased_loss_partial<<<kBlocks, kTPB, 0, stream>>>(x, tgt, partials);
  biased_loss_final<<<1, 32, 0, stream>>>(partials, out);
}